// PyramidROIAlignLayer_23390391894691
// MI455X (gfx1250) — compile-verified
//
#include <hip/hip_runtime.h>
#include <hip/hip_bf16.h>

// PyramidROIAlign for MI455X (gfx1250, wave32).
// Bandwidth-bound op (~0.5 GB traffic -> ~25us at 23.3 TB/s). Strategy:
//  - one 256-thread block (8 wave32) per ROI
//  - horizontal bilinear pass: coalesced float4 gathers (32 lanes x 4 = 128ch)
//    into LDS Hrow[16 slots][7*128], x-validity folded into weights
//  - vertical pass = 16x16x896 GEMM via V_WMMA_F32_16X16X4_F32:
//    out[i,(j,c)] = sum_r Wy[i,r] * Hrow[r,(j,c)], Wy carries y-weights/validity
//  - channels processed in 2 chunks of 128 to bound LDS; chunk-1 prefetched.

#define POOLN   7
#define NBOX    1000
#define NBATCH  2
#define NCH     256
#define CCHUNK  128
#define NCOLS   (POOLN * CCHUNK)      // 896
#define LDS_STRIDE 904                // 896 + 8 pad -> conflict-free B loads
#define NSLOT   14                    // 2 source rows per output row

typedef float v2f __attribute__((ext_vector_type(2)));
typedef float v8f __attribute__((ext_vector_type(8)));

__global__ __launch_bounds__(256)
void pyramid_roi_align_kernel(const float* __restrict__ rois,
                              const float* __restrict__ p2,
                              const float* __restrict__ p3,
                              const float* __restrict__ p4,
                              const float* __restrict__ p5,
                              float* __restrict__ out)
{
    __shared__ float hrow[16 * LDS_STRIDE];

    const int roi  = blockIdx.x;          // 0..1999
    const int tid  = threadIdx.x;
    const int lane = tid & 31;
    const int wave = tid >> 5;            // 0..7
    const int bidx = roi / NBOX;

    // ---- per-ROI uniform scalars (every thread computes the same) ----
    const float y1 = rois[roi * 4 + 0];
    const float x1 = rois[roi * 4 + 1];
    const float y2 = rois[roi * 4 + 2];
    const float x2 = rois[roi * 4 + 3];
    const float bh = y2 - y1, bw = x2 - x1;

    // roi_level = clip(4 + round(log2(sqrt(h*w) * 1024/224)), 2, 5)
    float lvlf = log2f(sqrtf(fmaxf(bh * bw, 1e-30f)) * (1024.0f / 224.0f));
    int lvl = 4 + (int)rintf(lvlf);       // rintf == round-half-even (jnp.round)
    lvl = min(max(lvl, 2), 5);

    const float* feat; int H;
    if      (lvl == 2) { feat = p2; H = 256; }
    else if (lvl == 3) { feat = p3; H = 128; }
    else if (lvl == 4) { feat = p4; H = 64;  }
    else               { feat = p5; H = 32;  }
    const int W = H;
    const float* fb = feat + (size_t)bidx * H * W * NCH;

    const float ybase = y1 * (float)(H - 1);
    const float ystep = bh * (float)(H - 1) * (1.0f / 6.0f);
    const float xbase = x1 * (float)(W - 1);
    const float xstep = bw * (float)(W - 1) * (1.0f / 6.0f);

    // ---- zero K-padding slots 14,15 of Hrow (so WMMA K padding adds 0) ----
    for (int k = tid; k < 2 * LDS_STRIDE; k += 256)
        hrow[NSLOT * LDS_STRIDE + k] = 0.0f;

    // ---- build A fragment (Wy, 16x16 padded; y-validity folded in) ----
    // A-matrix 16x4 layout (ISA 7.12.2): lanes 0-15 hold M=lane, VGPR0/1 give
    // K = khalf, khalf+1 with khalf = 0 (lanes 0-15) or 2 (lanes 16-31).
    const int m16   = lane & 15;
    const int khalf = (lane >> 4) << 1;

    auto Aval = [&](int m, int k) -> float {
        if (m > 6 || k >= NSLOT) return 0.0f;
        if ((k >> 1) != m)       return 0.0f;
        float ys  = ybase + (float)m * ystep;
        float y0f = floorf(ys);
        if (!(ys >= 0.0f && ys <= (float)(H - 1))) return 0.0f; // y-invalid
        float wy = ys - y0f;
        return (k & 1) ? wy : (1.0f - wy);
    };

    v2f afrag[4];
#pragma unroll
    for (int kc = 0; kc < 4; ++kc) {
        afrag[kc].x = Aval(m16, kc * 4 + khalf + 0);
        afrag[kc].y = Aval(m16, kc * 4 + khalf + 1);
    }

    // ---- channel chunks ----
    for (int cb = 0; cb < NCH; cb += CCHUNK) {
        __syncthreads();   // previous chunk's WMMA reads done before overwrite

        // ===== horizontal pass: 98 (slot,row-col) pairs over 8 waves =====
        for (int p = wave; p < NSLOT * POOLN; p += 8) {
            const int r = p / POOLN;       // slot 0..13
            const int j = p % POOLN;       // output column
            const int i = r >> 1;

            float ys  = ybase + (float)i * ystep;
            float y0f = floorf(ys);
            int   yq  = (int)y0f + (r & 1);
            yq = min(max(yq, 0), H - 1);

            float xs  = xbase + (float)j * xstep;
            float x0f = floorf(xs);
            bool  xv  = (xs >= 0.0f) && (xs <= (float)(W - 1));
            int   xa  = min(max((int)x0f,     0), W - 1);
            int   xb  = min(max((int)x0f + 1, 0), W - 1);
            float wx  = xs - x0f;
            float w0  = xv ? (1.0f - wx) : 0.0f;
            float w1  = xv ? wx          : 0.0f;

            const float* r0 = fb + ((size_t)yq * W + xa) * NCH + cb;
            const float* r1 = fb + ((size_t)yq * W + xb) * NCH + cb;
            float4 v0 = ((const float4*)r0)[lane];   // ch cb + lane*4 .. +3
            float4 v1 = ((const float4*)r1)[lane];

            if (cb == 0) {                 // prefetch next channel chunk
                __builtin_prefetch(r0 + CCHUNK + lane * 4, 0, 3);
                __builtin_prefetch(r1 + CCHUNK + lane * 4, 0, 3);
            }

            float4 o;
            o.x = v0.x * w0 + v1.x * w1;
            o.y = v0.y * w0 + v1.y * w1;
            o.z = v0.z * w0 + v1.z * w1;
            o.w = v0.w * w0 + v1.w * w1;
            *(float4*)&hrow[r * LDS_STRIDE + j * CCHUNK + lane * 4] = o;
        }

        __syncthreads();

        // ===== vertical pass: 16x16x(4x4) WMMA over 56 N-tiles =====
        for (int t = wave; t < NCOLS / 16; t += 8) {
            const int ncol = t * 16;
            v8f acc = {};
#pragma unroll
            for (int kc = 0; kc < 4; ++kc) {
                const int kb = kc * 4 + khalf;
                // B-matrix 4x16: row K striped across lanes within a VGPR,
                // lane halves take K = kb / kb+1 (mirrors the A layout).
                v2f bfrag;
                bfrag.x = hrow[(kb + 0) * LDS_STRIDE + ncol + m16];
                bfrag.y = hrow[(kb + 1) * LDS_STRIDE + ncol + m16];
                acc = __builtin_amdgcn_wmma_f32_16x16x4_f32(
                    /*neg_a=*/false, afrag[kc],
                    /*neg_b=*/false, bfrag,
                    /*c_mod=*/(short)0, acc,
                    /*reuse_a=*/false, /*reuse_b=*/false);
            }
            // D layout: lanes 0-15 hold N=lane, VGPR v -> M=v. Only rows
            // M=0..6 (POOL=7) are real; lanes 16-31 hold M>=8 (all padding).
            if (lane < 16) {
                const int nl = ncol + lane;          // 0..895
                const int j  = nl / CCHUNK;
                const int c  = cb + (nl % CCHUNK);
                float* op = out + (size_t)roi * (POOLN * POOLN * NCH)
                                + (size_t)j * NCH + c;
#pragma unroll
                for (int v = 0; v < POOLN; ++v)
                    op[(size_t)v * (POOLN * NCH)] = acc[v];
            }
        }
    }
}

extern "C" void kernel_launch(void* const* d_in, const int* in_sizes, int n_in,
                              void* d_out, int out_size, void* d_ws, size_t ws_size,
                              hipStream_t stream) {
    (void)in_sizes; (void)n_in; (void)d_ws; (void)ws_size; (void)out_size;
    const float* rois = (const float*)d_in[0];
    const float* p2   = (const float*)d_in[1];
    const float* p3   = (const float*)d_in[2];
    const float* p4   = (const float*)d_in[3];
    const float* p5   = (const float*)d_in[4];
    float* out = (float*)d_out;

    dim3 grid(NBATCH * NBOX);   // one block per ROI
    dim3 block(256);            // 8 wave32
    pyramid_roi_align_kernel<<<grid, block, 0, stream>>>(rois, p2, p3, p4, p5, out);
}